// DILATEloss_35476429865779
// MI455X (gfx1250) — compile-verified
//
#include <hip/hip_runtime.h>
#include <math.h>

typedef __attribute__((ext_vector_type(2))) float v2f;
typedef __attribute__((ext_vector_type(8))) float v8f;

#define Ln 256
#define Bn 64
#define GAMMA_F 0.01f
#define INV_G 100.0f
#define INF_V 1e8f

__device__ __forceinline__ float softmin3(float a, float b, float c) {
    float m = fminf(a, fminf(b, c));
    float s = expf((m - a) * INV_G) + expf((m - b) * INV_G) + expf((m - c) * INV_G);
    return m - GAMMA_F * logf(s);
}

// ---------------- Forward soft-DTW: one block per batch, anti-diagonal wavefront
__global__ void __launch_bounds__(256)
dilate_fwd_kernel(const float* __restrict__ inp, const float* __restrict__ tgt,
                  float* __restrict__ Rg, float* __restrict__ vals) {
    __shared__ float t_s[Ln], x_s[Ln];
    __shared__ float diag[3][Ln + 1];   // slot j of diag(d) holds R[d-j, j]
    const int b = blockIdx.x, tid = threadIdx.x;
    const int j = tid + 1;              // 1-based column
    t_s[tid] = tgt[b * Ln + tid];
    x_s[tid] = inp[b * Ln + tid];
    diag[0][tid] = INF_V; diag[1][tid] = INF_V; diag[2][tid] = INF_V;
    if (tid == 0) {
        diag[0][Ln] = INF_V; diag[1][Ln] = INF_V; diag[2][Ln] = INF_V;
        diag[0][0] = 0.0f;              // R[0,0] = 0
    }
    __syncthreads();
    const float xj = x_s[tid];
    float rlast = 0.0f;
    float* __restrict__ rb = Rg + (size_t)b * Ln * Ln;
    for (int d = 2; d <= 2 * Ln; ++d) {
        float*       cur   = diag[d % 3];
        const float* prev  = diag[(d + 2) % 3];  // diag d-1
        const float* prev2 = diag[(d + 1) % 3];  // diag d-2
        const int i = d - j;
        if (i >= 1 && i <= Ln) {
            float du = t_s[i - 1] - xj; du *= du;                 // D[i,j]
            float r = du + softmin3(prev2[j - 1], prev[j], prev[j - 1]);
            cur[j] = r;
            rb[(i - 1) * Ln + (j - 1)] = r;
            rlast = r;
        }
        if (tid == 0) cur[0] = INF_V;   // R[d,0]
        if (j == d)   cur[j] = INF_V;   // R[0,j]
        __syncthreads();
    }
    if (tid == Ln - 1) vals[b] = rlast; // R[L,L]
}

// ---------------- Backward: E[i,j] recursion over reverse anti-diagonals
__global__ void __launch_bounds__(256)
dilate_bwd_kernel(const float* __restrict__ inp, const float* __restrict__ tgt,
                  const float* __restrict__ Rg, float* __restrict__ Eg) {
    __shared__ float t_s[Ln], x_s[Ln];
    __shared__ float ed[3][Ln + 1];     // E diagonals
    __shared__ float rd[3][Ln + 1];     // R diagonals (cached, 1 global load/cell)
    const int b = blockIdx.x, tid = threadIdx.x;
    const int j = tid + 1;
    t_s[tid] = tgt[b * Ln + tid];
    x_s[tid] = inp[b * Ln + tid];
    #pragma unroll
    for (int v = 0; v < 3; ++v) { ed[v][tid] = 0.0f; rd[v][tid] = 0.0f; }
    if (tid == 0) {
        #pragma unroll
        for (int v = 0; v < 3; ++v) { ed[v][Ln] = 0.0f; rd[v][Ln] = 0.0f; }
    }
    __syncthreads();
    const float xj = x_s[tid];
    const float* __restrict__ rb = Rg + (size_t)b * Ln * Ln;
    float* __restrict__ eb = Eg + (size_t)b * Ln * Ln;
    for (int d = 2 * Ln; d >= 2; --d) {
        float*       ecur = ed[d % 3];
        float*       rcur = rd[d % 3];
        const float* ep   = ed[(d + 1) % 3];  // diag d+1
        const float* ep2  = ed[(d + 2) % 3];  // diag d+2
        const float* rp   = rd[(d + 1) % 3];
        const float* rp2  = rd[(d + 2) % 3];
        const int i = d - j;
        if (i >= 1 && i <= Ln) {
            const float rij = rb[(i - 1) * Ln + (j - 1)];
            rcur[j] = rij;
            float e = 0.0f;
            if (i < Ln) {                                    // successor (i+1, j)
                float dd = t_s[i] - xj; dd *= dd;
                e += ep[j] * expf((rp[j] - rij - dd) * INV_G);
            }
            if (j < Ln) {                                    // successor (i, j+1)
                float dd = t_s[i - 1] - x_s[j]; dd *= dd;
                e += ep[j + 1] * expf((rp[j + 1] - rij - dd) * INV_G);
            }
            if (i < Ln && j < Ln) {                          // successor (i+1, j+1)
                float dd = t_s[i] - x_s[j]; dd *= dd;
                e += ep2[j + 1] * expf((rp2[j + 1] - rij - dd) * INV_G);
            } else if (i == Ln && j == Ln) {
                e += 1.0f;                                   // E[L+1,L+1] seed
            }
            ecur[j] = e;
            eb[(i - 1) * Ln + (j - 1)] = e;
        }
        __syncthreads();
    }
}

// ---------------- WMMA contraction:  tsum[b] = sum_ij E[i,j] * (i-j)^2 / L^2
// trace(E_tile x Omega_tile^T) == elementwise sum of E_tile .* Omega_tile.
// Each wave accumulates 32 tiles x 4 K-chunks into ONE f32 accumulator,
// extracting the diagonal once at the end.
__global__ void __launch_bounds__(256)
dilate_wmma_reduce_kernel(const float* __restrict__ Eg, float* __restrict__ tsum) {
    __shared__ float red[256];
    const int b = blockIdx.x, tid = threadIdx.x;
    const int wave = tid >> 5, lane = tid & 31;
    const float* __restrict__ eb = Eg + (size_t)b * Ln * Ln;
    const int m    = lane & 15;          // A: row M; B: col N (same lane slot)
    const int koff = (lane >> 4) * 2;    // K pair offset per ISA 16x4 f32 layout
    const float invLL = 1.0f / (float)(Ln * Ln);
    v8f acc = {};
    for (int t = wave * 32; t < wave * 32 + 32; ++t) {   // 256 tiles / 8 waves
        const int ti = t >> 4, tj = t & 15;
        const int gi = ti * 16 + m;                       // global row index
        const float* arow = eb + gi * Ln + tj * 16;
        #pragma unroll
        for (int kk = 0; kk < 4; ++kk) {
            const int c0 = kk * 4 + koff;
            v2f a, w;
            a.x = arow[c0];
            a.y = arow[c0 + 1];
            const int gj0 = tj * 16 + c0;                 // global col index
            float d0 = (float)(gi - gj0);
            float d1 = (float)(gi - gj0 - 1);
            w.x = d0 * d0 * invLL;                        // Omega^T element
            w.y = d1 * d1 * invLL;
            acc = __builtin_amdgcn_wmma_f32_16x16x4_f32(
                false, a, false, w, (short)0, acc, false, false);
        }
    }
    // D[M][N]: M = v + (lane>=16)*8, N = lane&15 ; keep diagonal M==N
    float tmp[8];
    #pragma unroll
    for (int v = 0; v < 8; ++v) tmp[v] = acc[v];
    const int vd = (lane & 15) - (lane >> 4) * 8;
    red[tid] = (vd >= 0 && vd < 8) ? tmp[vd] : 0.0f;
    __syncthreads();
    for (int s = 128; s > 0; s >>= 1) {
        if (tid < s) red[tid] += red[tid + s];
        __syncthreads();
    }
    if (tid == 0) tsum[b] = red[0];
}

// ---------------- Final scalar:  0.5*mean(vals) + 0.5*mean(tsum)
__global__ void dilate_final_kernel(const float* __restrict__ vals,
                                    const float* __restrict__ tsum,
                                    float* __restrict__ out) {
    __shared__ float s1[Bn], s2[Bn];
    const int tid = threadIdx.x;
    s1[tid] = vals[tid];
    s2[tid] = tsum[tid];
    __syncthreads();
    for (int s = Bn / 2; s > 0; s >>= 1) {
        if (tid < s) { s1[tid] += s1[tid + s]; s2[tid] += s2[tid + s]; }
        __syncthreads();
    }
    if (tid == 0)
        out[0] = 0.5f * (s1[0] / (float)Bn) + 0.5f * (s2[0] / (float)Bn);
}

extern "C" void kernel_launch(void* const* d_in, const int* in_sizes, int n_in,
                              void* d_out, int out_size, void* d_ws, size_t ws_size,
                              hipStream_t stream) {
    const float* inp = (const float*)d_in[0];   // [64,256,1]
    const float* tgt = (const float*)d_in[1];   // [64,256,1]
    float* out = (float*)d_out;                 // [1]

    float* ws   = (float*)d_ws;
    float* Rg   = ws;                                   // 64*256*256
    float* Eg   = ws + (size_t)Bn * Ln * Ln;            // 64*256*256
    float* vals = Eg + (size_t)Bn * Ln * Ln;            // 64
    float* tsum = vals + Bn;                            // 64

    dilate_fwd_kernel<<<Bn, 256, 0, stream>>>(inp, tgt, Rg, vals);
    dilate_bwd_kernel<<<Bn, 256, 0, stream>>>(inp, tgt, Rg, Eg);
    dilate_wmma_reduce_kernel<<<Bn, 256, 0, stream>>>(Eg, tsum);
    dilate_final_kernel<<<1, Bn, 0, stream>>>(vals, tsum, out);
}